// Transformer_10239202034068
// MI455X (gfx1250) — compile-verified
//
#include <hip/hip_runtime.h>
#include <hip/hip_bf16.h>
#include <math.h>

typedef __attribute__((ext_vector_type(2))) float v2f;
typedef __attribute__((ext_vector_type(4))) float v4f;
typedef __attribute__((ext_vector_type(8))) float v8f;

#define D_MODEL 4096
#define D_FF    512
#define LAYER   3          // only layer 3 affects the output (see analysis)
#define EPSF    1e-6f

// ---------------------------------------------------------------------------
// Column-access GEMV (e = W_e^T @ src), split over M into partials.
// partial[s*D + d] = sum_{m in chunk s} src[m] * W[m*D + d]
// ---------------------------------------------------------------------------
__global__ void colgemv_partial(const float* __restrict__ W,
                                const float* __restrict__ x,
                                float* __restrict__ partial,
                                int D, int Mc) {
  const int d = blockIdx.x * blockDim.x + threadIdx.x;   // output column
  const int s = blockIdx.y;                              // m-chunk
  const int m0 = s * Mc;
  const float* Wp = W + (size_t)m0 * D + d;
  float acc = 0.f;
#pragma unroll 8
  for (int m = 0; m < Mc; ++m)
    acc = fmaf(x[m0 + m], Wp[(size_t)m * D], acc);
  partial[(size_t)s * D + d] = acc;
}

// ---------------------------------------------------------------------------
// Row-dot GEMV tile via V_WMMA_F32_16X16X4_F32.
// One wave -> 16 output rows; split-K over blockIdx.y chunks of Kc columns.
// A slots: lanes 0-15 carry (row r, cols k..k+1); lanes 16-31 carry
// (row r, cols k+4..k+5) for WMMA#1 (and +2/+6 for WMMA#2). B carries the
// matching x value in the matching (VGPR, lane-half) slot, broadcast over N,
// so every column of C equals the 16 row dot-products.
// ---------------------------------------------------------------------------
__global__ void rowdot_wmma(const float* __restrict__ W,
                            const float* __restrict__ x,
                            float* __restrict__ partial,
                            int nrows, int ncols, int Kc) {
  const int lane = threadIdx.x & 31;
  const int wave = (blockIdx.x * blockDim.x + threadIdx.x) >> 5;
  const int row0 = wave * 16;
  const int half = lane >> 4;     // 0: K-slots 0,1   1: K-slots 2,3
  const int r    = lane & 15;     // row within 16-row tile
  const int kb   = blockIdx.y * Kc;

  const float* Wrow = W + (size_t)(row0 + r) * ncols + kb + half * 4;
  const float* xp   = x + kb + half * 4;

  v8f acc = {};
#pragma unroll 8
  for (int k = 0; k < Kc; k += 8) {
    v4f w  = *(const v4f*)(Wrow + k);   // 4 consecutive cols of my row
    v4f xv = *(const v4f*)(xp + k);     // matching 4 x values
    v2f a0 = { w.x, w.y };
    v2f b0 = { xv.x, xv.y };
    acc = __builtin_amdgcn_wmma_f32_16x16x4_f32(
        false, a0, false, b0, (short)0, acc, false, false);
    v2f a1 = { w.z, w.w };
    v2f b1 = { xv.z, xv.w };
    acc = __builtin_amdgcn_wmma_f32_16x16x4_f32(
        false, a1, false, b1, (short)0, acc, false, false);
  }

  // C layout: VGPR i, lane-half h  ->  M = i + 8*h (all N columns identical).
  if (r == 0) {
    float* out = partial + (size_t)blockIdx.y * nrows + row0 + half * 8;
#pragma unroll
    for (int i = 0; i < 8; ++i) out[i] = acc[i];
  }
}

// ---------------------------------------------------------------------------
// Deterministic fixed-order split-K reduction + bias (+ optional relu).
// ---------------------------------------------------------------------------
__global__ void reduce_partials(const float* __restrict__ partial,
                                const float* __restrict__ bias,
                                float* __restrict__ y,
                                int n, int S, int relu) {
  const int j = blockIdx.x * blockDim.x + threadIdx.x;
  if (j >= n) return;
  float acc = bias[j];
  for (int s = 0; s < S; ++s) acc += partial[(size_t)s * n + j];
  if (relu) acc = fmaxf(acc, 0.f);
  y[j] = acc;
}

// ---------------------------------------------------------------------------
// Block-wide sum (blockDim must be exactly 1024 = 32 wave32s).
// ---------------------------------------------------------------------------
__device__ __forceinline__ float blockReduceSum(float v, float* sred) {
#pragma unroll
  for (int off = 16; off > 0; off >>= 1) v += __shfl_xor(v, off, 32);
  const int lane = threadIdx.x & 31;
  const int wid  = threadIdx.x >> 5;
  __syncthreads();
  if (lane == 0) sred[wid] = v;
  __syncthreads();
  float t = sred[lane];            // 32 partials, every wave reduces them
#pragma unroll
  for (int off = 16; off > 0; off >>= 1) t += __shfl_xor(t, off, 32);
  return t;
}

// ---------------------------------------------------------------------------
// y = alpha * (x - mu) / (sqrt(var_unbiased) + eps) + beta,  x = in1 (+ in2).
// Optionally stores x to xsave. Single block of 1024 threads, N = 4096.
// ---------------------------------------------------------------------------
__global__ void norm_fused(const float* __restrict__ in1,
                           const float* __restrict__ in2,
                           const float* __restrict__ alpha,
                           const float* __restrict__ beta,
                           float* __restrict__ xsave,
                           float* __restrict__ y) {
  __shared__ float sred[32];
  const int tid = threadIdx.x;
  float vals[4];
  float lsum = 0.f;
#pragma unroll
  for (int i = 0; i < 4; ++i) {
    int idx = tid + i * 1024;
    float v = in1[idx];
    if (in2) v += in2[idx];
    vals[i] = v;
    lsum += v;
  }
  const float mu = blockReduceSum(lsum, sred) * (1.f / (float)D_MODEL);
  float lvar = 0.f;
#pragma unroll
  for (int i = 0; i < 4; ++i) { float d = vals[i] - mu; lvar += d * d; }
  const float var = blockReduceSum(lvar, sred) * (1.f / (float)(D_MODEL - 1));
  const float s = 1.f / (sqrtf(var) + EPSF);
#pragma unroll
  for (int i = 0; i < 4; ++i) {
    int idx = tid + i * 1024;
    if (xsave) xsave[idx] = vals[i];
    y[idx] = alpha[idx] * (vals[i] - mu) * s + beta[idx];
  }
}

// ---------------------------------------------------------------------------
// Final: x = x1 + ff; xf = norm(x); out = dot(xf, Wout) + bout.
// ---------------------------------------------------------------------------
__global__ void final_norm_dot(const float* __restrict__ x1,
                               const float* __restrict__ ff,
                               const float* __restrict__ alpha,
                               const float* __restrict__ beta,
                               const float* __restrict__ wout,
                               const float* __restrict__ bout,
                               float* __restrict__ out) {
  __shared__ float sred[32];
  const int tid = threadIdx.x;
  float vals[4];
  float lsum = 0.f;
#pragma unroll
  for (int i = 0; i < 4; ++i) {
    int idx = tid + i * 1024;
    float v = x1[idx] + ff[idx];
    vals[i] = v;
    lsum += v;
  }
  const float mu = blockReduceSum(lsum, sred) * (1.f / (float)D_MODEL);
  float lvar = 0.f;
#pragma unroll
  for (int i = 0; i < 4; ++i) { float d = vals[i] - mu; lvar += d * d; }
  const float var = blockReduceSum(lvar, sred) * (1.f / (float)(D_MODEL - 1));
  const float s = 1.f / (sqrtf(var) + EPSF);
  float ldot = 0.f;
#pragma unroll
  for (int i = 0; i < 4; ++i) {
    int idx = tid + i * 1024;
    float xf = alpha[idx] * (vals[i] - mu) * s + beta[idx];
    ldot = fmaf(xf, wout[idx], ldot);
  }
  const float dot = blockReduceSum(ldot, sred);
  if (tid == 0) out[0] = dot + bout[0];
}

// ---------------------------------------------------------------------------
extern "C" void kernel_launch(void* const* d_in, const int* in_sizes, int n_in,
                              void* d_out, int out_size, void* d_ws, size_t ws_size,
                              hipStream_t stream) {
  (void)in_sizes; (void)n_in; (void)out_size; (void)ws_size;

  const float* src  = (const float*)d_in[0];
  const float* W_e  = (const float*)d_in[1];
  const float* b_e  = (const float*)d_in[2];
  const float* ln1a = (const float*)d_in[3]  + (size_t)LAYER * D_MODEL;
  const float* ln1b = (const float*)d_in[4]  + (size_t)LAYER * D_MODEL;
  // d_in[5..8] = Wq,bq,Wk,bk: provably dead (softmax over length-1 axis == 1)
  const float* Wv   = (const float*)d_in[9]  + (size_t)LAYER * D_MODEL * D_MODEL;
  const float* bv   = (const float*)d_in[10] + (size_t)LAYER * D_MODEL;
  const float* Wo   = (const float*)d_in[11] + (size_t)LAYER * D_MODEL * D_MODEL;
  const float* bo   = (const float*)d_in[12] + (size_t)LAYER * D_MODEL;
  const float* ln2a = (const float*)d_in[13] + (size_t)LAYER * D_MODEL;
  const float* ln2b = (const float*)d_in[14] + (size_t)LAYER * D_MODEL;
  const float* W1   = (const float*)d_in[15] + (size_t)LAYER * D_FF * D_MODEL;
  const float* b1   = (const float*)d_in[16] + (size_t)LAYER * D_FF;
  const float* W2   = (const float*)d_in[17] + (size_t)LAYER * D_MODEL * D_FF;
  const float* b2   = (const float*)d_in[18] + (size_t)LAYER * D_MODEL;
  const float* lnfa = (const float*)d_in[19];
  const float* lnfb = (const float*)d_in[20];
  const float* Wout = (const float*)d_in[21];
  const float* bout = (const float*)d_in[22];

  float* ws   = (float*)d_ws;
  float* e    = ws + 0 * D_MODEL;
  float* x2a  = ws + 1 * D_MODEL;
  float* vvec = ws + 2 * D_MODEL;
  float* ovec = ws + 3 * D_MODEL;
  float* x1   = ws + 4 * D_MODEL;
  float* x2b  = ws + 5 * D_MODEL;
  float* hvec = ws + 6 * D_MODEL;  // only 512 used
  float* ffv  = ws + 7 * D_MODEL;
  float* part = ws + 8 * D_MODEL;  // up to 32*4096 floats

  // 1) e = src @ W_e + b_e   (split-M column GEMV, 32 chunks of 128 rows)
  colgemv_partial<<<dim3(D_MODEL / 256, 32), 256, 0, stream>>>(W_e, src, part, D_MODEL, 128);
  reduce_partials<<<D_MODEL / 256, 256, 0, stream>>>(part, b_e, e, D_MODEL, 32, 0);

  // 2) x2a = norm(e, ln1[3])
  norm_fused<<<1, 1024, 0, stream>>>(e, nullptr, ln1a, ln1b, nullptr, x2a);

  // 3) v = Wv3 @ x2a + bv3   (attn == v since softmax over seq-1 is identity)
  rowdot_wmma<<<dim3(32, 16), 256, 0, stream>>>(Wv, x2a, part, D_MODEL, D_MODEL, 256);
  reduce_partials<<<D_MODEL / 256, 256, 0, stream>>>(part, bv, vvec, D_MODEL, 16, 0);

  // 4) o = Wo3 @ v + bo3
  rowdot_wmma<<<dim3(32, 16), 256, 0, stream>>>(Wo, vvec, part, D_MODEL, D_MODEL, 256);
  reduce_partials<<<D_MODEL / 256, 256, 0, stream>>>(part, bo, ovec, D_MODEL, 16, 0);

  // 5) x1 = e + o ; x2b = norm(x1, ln2[3])
  norm_fused<<<1, 1024, 0, stream>>>(e, ovec, ln2a, ln2b, x1, x2b);

  // 6) h = relu(W1_3 @ x2b + b1_3)   (512 x 4096)
  rowdot_wmma<<<dim3(4, 16), 256, 0, stream>>>(W1, x2b, part, D_FF, D_MODEL, 256);
  reduce_partials<<<2, 256, 0, stream>>>(part, b1, hvec, D_FF, 16, 1);

  // 7) ff = W2_3 @ h + b2_3   (4096 x 512)
  rowdot_wmma<<<dim3(32, 4), 256, 0, stream>>>(W2, hvec, part, D_MODEL, D_FF, 128);
  reduce_partials<<<D_MODEL / 256, 256, 0, stream>>>(part, b2, ffv, D_MODEL, 4, 0);

  // 8) out = norm(x1 + ff, lnf) . Wout + bout
  final_norm_dot<<<1, 1024, 0, stream>>>(x1, ffv, lnfa, lnfb, Wout, bout, (float*)d_out);
}